// Additive_84653805404421
// MI455X (gfx1250) — compile-verified
//
#include <hip/hip_runtime.h>

typedef __attribute__((ext_vector_type(2))) float v2f;
typedef __attribute__((ext_vector_type(4))) float v4f;
typedef __attribute__((ext_vector_type(8))) float v8f;

// Exact parameter type of __builtin_amdgcn_global_load_async_to_lds_b128 per
// hipcc diagnostic: "__attribute__((__vector_size__(4 * sizeof(int)))) int *"
// in addrspace(1) (printed as __device__) / addrspace(3).
typedef int v4i __attribute__((vector_size(16)));
typedef __attribute__((address_space(1))) v4i* gv4i_p;
typedef __attribute__((address_space(3))) v4i* lv4i_p;

#define DIM 512
#define NR  1024

#if __has_builtin(__builtin_amdgcn_global_load_async_to_lds_b128)
#define HAVE_ASYNC_LDS 1
#else
#define HAVE_ASYNC_LDS 0
#warning "global_load_async_to_lds_b128 builtin unavailable - VGPR-staged LDS fill"
#endif

__device__ __forceinline__ void wait_asynccnt0() {
#if __has_builtin(__builtin_amdgcn_s_wait_asynccnt)
    __builtin_amdgcn_s_wait_asynccnt(0);
#else
    asm volatile("s_wait_asynccnt 0x0" ::: "memory");
#endif
}

// ---------------------------------------------------------------------------
// Stage 1: H = A @ W tile GEMM using V_WMMA_F32_16X16X4_F32 (full f32 precision,
// matches the f32 reference; K accumulated in f32).
// One wave computes one 16x16 tile of H. blockIdx.y: 0 -> hx = x@W1x,
// 1 -> hy = y@W1y (W1y lives at W1 + DIM*DIM, row-major (2*DIM, DIM)).
//
// ISA layouts (CDNA5 ISA §7.12.2, wave32):
//   A (16x4, MxK):  lane&15 = M; lanes 0-15 hold K=0(V0),1(V1); lanes 16-31 K=2,3
//   B (4x16, KxN):  lane&15 = N; lanes 0-15 hold K=0(V0),1(V1); lanes 16-31 K=2,3
//   C/D (16x16):    VGPR v: lanes 0-15 -> M=v, lanes 16-31 -> M=v+8; N = lane&15
// ---------------------------------------------------------------------------
__global__ void __launch_bounds__(128)
wmma_h_gemm(const float* __restrict__ X, const float* __restrict__ Y,
            const float* __restrict__ W1,
            float* __restrict__ HX, float* __restrict__ HY)
{
    const int lane    = threadIdx.x & 31;
    const int wave    = threadIdx.x >> 5;
    const int tile    = blockIdx.x * 4 + wave;   // 0..2047
    const int rowTile = tile >> 5;               // 0..63  (1024/16 rows)
    const int colTile = tile & 31;               // 0..31  (512/16 cols)

    const float* A = blockIdx.y ? Y  : X;
    const float* W = W1 + (blockIdx.y ? (size_t)DIM * DIM : 0);
    float*       H = blockIdx.y ? HY : HX;

    const int m  = lane & 15;
    const int kg = (lane >> 4) << 1;             // 0 for lanes 0-15, 2 for 16-31

    const float* aPtr = A + (size_t)(rowTile * 16 + m) * DIM + kg; // A[m][k0+kg..+1]
    const float* bPtr = W + (size_t)kg * DIM + colTile * 16 + m;   // W[k0+kg][n]

    v8f acc = {};

#if __has_builtin(__builtin_amdgcn_wmma_f32_16x16x4_f32)
    #pragma unroll 4
    for (int k0 = 0; k0 < DIM; k0 += 4) {
        v2f a = *(const v2f*)(aPtr + k0);        // contiguous K pair, 8B aligned
        v2f b;
        b.x = bPtr[(size_t)k0 * DIM];
        b.y = bPtr[(size_t)k0 * DIM + DIM];
        acc = __builtin_amdgcn_wmma_f32_16x16x4_f32(
            /*neg_a=*/false, a, /*neg_b=*/false, b,
            /*c_mod=*/(short)0, acc, /*reuse_a=*/false, /*reuse_b=*/false);
    }
#else
    #warning "wmma_f32_16x16x4_f32 builtin unavailable on this toolchain - scalar fallback"
    for (int k0 = 0; k0 < DIM; k0 += 4) {
        v2f a = *(const v2f*)(aPtr + k0);
        v2f b;
        b.x = bPtr[(size_t)k0 * DIM];
        b.y = bPtr[(size_t)k0 * DIM + DIM];
        acc[0] = fmaf(a.x, b.x, fmaf(a.y, b.y, acc[0]));
    }
#endif

    float*    outP  = H + (size_t)colTile * 16 + (lane & 15);
    const int mBase = rowTile * 16 + ((lane >> 4) << 3);
    #pragma unroll
    for (int v = 0; v < 8; ++v)
        outP[(size_t)(mBase + v) * DIM] = acc[v];
}

// ---------------------------------------------------------------------------
// Stage 2 (non-pairwise): out[i,j] = sum_d relu(hx[i,d] + hy[j,d]) * w2[d]
// 32x32 output tile per 256-thread block; d staged through LDS in chunks of 128.
// LDS fill uses GLOBAL_LOAD_ASYNC_TO_LDS_B128 (ASYNCcnt-tracked DMA, no VGPR
// round-trip) when the builtin exists. Each thread owns 1 i and 4 consecutive j.
// Rows padded by 4 floats: float4-aligned, reduced bank conflicts.
// ---------------------------------------------------------------------------
#define TI  32
#define TJ  32
#define DC  128
#define RS  (DC + 4)

__global__ void __launch_bounds__(256)
pair_reduce(const float* __restrict__ HX, const float* __restrict__ HY,
            const float* __restrict__ W2, float* __restrict__ OUT)
{
    __shared__ float sX[TI * RS];
    __shared__ float sY[TJ * RS];
    __shared__ float sW[DC];

    const int tid   = threadIdx.x;
    const int i     = tid >> 3;          // 0..31
    const int j4    = (tid & 7) << 2;    // 0,4,...,28
    const int iBase = blockIdx.y * TI;
    const int jBase = blockIdx.x * TJ;

    float acc0 = 0.f, acc1 = 0.f, acc2 = 0.f, acc3 = 0.f;

    for (int d0 = 0; d0 < DIM; d0 += DC) {
        // Cooperative fill: 32 rows x 128 floats = 1024 16B transfers per matrix.
#if HAVE_ASYNC_LDS
        #pragma unroll
        for (int t = 0; t < 4; ++t) {
            int idx = tid + t * 256;                 // 0..1023
            int r   = idx >> 5;                      // row 0..31
            int c   = (idx & 31) << 2;               // col 0,4,...,124
            __builtin_amdgcn_global_load_async_to_lds_b128(
                (gv4i_p)(HX + (size_t)(iBase + r) * DIM + d0 + c),
                (lv4i_p)(sX + r * RS + c),
                /*offset=*/0, /*cpol=*/0);
            __builtin_amdgcn_global_load_async_to_lds_b128(
                (gv4i_p)(HY + (size_t)(jBase + r) * DIM + d0 + c),
                (lv4i_p)(sY + r * RS + c),
                /*offset=*/0, /*cpol=*/0);
        }
        if (tid < DC) sW[tid] = W2[d0 + tid];        // overlaps with async DMA
        wait_asynccnt0();                            // drain this wave's copies
        __syncthreads();                             // publish to workgroup
#else
        #pragma unroll
        for (int t = 0; t < 4; ++t) {
            int idx = tid + t * 256;
            int r   = idx >> 5;
            int c   = (idx & 31) << 2;
            v4f vx = *(const v4f*)(HX + (size_t)(iBase + r) * DIM + d0 + c);
            v4f vy = *(const v4f*)(HY + (size_t)(jBase + r) * DIM + d0 + c);
            *(v4f*)(sX + r * RS + c) = vx;
            *(v4f*)(sY + r * RS + c) = vy;
        }
        if (tid < DC) sW[tid] = W2[d0 + tid];
        __syncthreads();
#endif

        const float* xr = sX + i * RS;
        const float* y0 = sY + (j4 + 0) * RS;
        const float* y1 = sY + (j4 + 1) * RS;
        const float* y2 = sY + (j4 + 2) * RS;
        const float* y3 = sY + (j4 + 3) * RS;
        #pragma unroll 8
        for (int dd = 0; dd < DC; ++dd) {
            float a = xr[dd];
            float w = sW[dd];
            acc0 = fmaf(fmaxf(a + y0[dd], 0.f), w, acc0);
            acc1 = fmaf(fmaxf(a + y1[dd], 0.f), w, acc1);
            acc2 = fmaf(fmaxf(a + y2[dd], 0.f), w, acc2);
            acc3 = fmaf(fmaxf(a + y3[dd], 0.f), w, acc3);
        }
        __syncthreads();
    }

    float* o = OUT + (size_t)(iBase + i) * NR + jBase + j4;
    o[0] = acc0; o[1] = acc1; o[2] = acc2; o[3] = acc3;
}

// ---------------------------------------------------------------------------
// Pairwise-mode fallback (out_size == N): out[i] = sum_d relu(hx[i,d]+hy[i,d])*w2[d]
// ---------------------------------------------------------------------------
__global__ void __launch_bounds__(256)
pair_rowwise(const float* __restrict__ HX, const float* __restrict__ HY,
             const float* __restrict__ W2, float* __restrict__ OUT)
{
    __shared__ float red[256];
    const int i = blockIdx.x;
    float s = 0.f;
    for (int d = threadIdx.x; d < DIM; d += 256)
        s = fmaf(fmaxf(HX[(size_t)i * DIM + d] + HY[(size_t)i * DIM + d], 0.f),
                 W2[d], s);
    red[threadIdx.x] = s;
    __syncthreads();
    for (int o = 128; o > 0; o >>= 1) {
        if (threadIdx.x < o) red[threadIdx.x] += red[threadIdx.x + o];
        __syncthreads();
    }
    if (threadIdx.x == 0) OUT[i] = red[0];
}

extern "C" void kernel_launch(void* const* d_in, const int* in_sizes, int n_in,
                              void* d_out, int out_size, void* d_ws, size_t ws_size,
                              hipStream_t stream)
{
    const float* x  = (const float*)d_in[0];
    const float* y  = (const float*)d_in[1];
    const float* W1 = (const float*)d_in[2];
    const float* W2 = (const float*)d_in[3];
    float* out = (float*)d_out;

    float* HX = (float*)d_ws;                 // 1024*512 f32 = 2 MB
    float* HY = HX + (size_t)NR * DIM;        // 2 MB more

    // Stage 1: both GEMMs in one launch; 2048 tiles per path, 4 waves/block.
    dim3 gemmGrid(512, 2, 1);
    wmma_h_gemm<<<gemmGrid, 128, 0, stream>>>(x, y, W1, HX, HY);

    if (out_size == NR * NR) {
        dim3 grid(NR / TJ, NR / TI, 1);
        pair_reduce<<<grid, 256, 0, stream>>>(HX, HY, W2, out);
    } else {
        pair_rowwise<<<NR, 256, 0, stream>>>(HX, HY, W2, out);
    }
}